// CorrClassLoss_43413529428716
// MI455X (gfx1250) — compile-verified
//
#include <hip/hip_runtime.h>
#include <hip/hip_bf16.h>

// Problem constants (match the reference)
#define BATCH  4
#define CCH    19
#define HH     512
#define WW     1024
#define HWSZ   (HH * WW)          // 524288
#define NPTS   5000
#define POINTS (BATCH * NPTS)     // 20000
#define TPB    256
#define NBLK   ((POINTS + TPB - 1) / TPB)   // 79
#define IGNORE_IDX 255

typedef float v2f __attribute__((ext_vector_type(2)));
typedef float v8f __attribute__((ext_vector_type(8)));

// ---- helpers -------------------------------------------------------------

__device__ __forceinline__ void load_point(const int* inds_ref, const int* inds_other,
                                           int b, int n,
                                           int& rx, int& ry, int& ox, int& oy, bool& valid) {
  const int* ir = inds_ref   + (size_t)b * 2 * NPTS;   // [2, N]: row0 = x, row1 = y
  const int* io = inds_other + (size_t)b * 2 * NPTS;
  rx = ir[n];  ry = ir[NPTS + n];
  ox = io[n];  oy = io[NPTS + n];
  valid = (rx >= 0) & (rx < WW) & (ry >= 0) & (ry < HH) &
          (ox >= 0) & (ox < WW) & (oy >= 0) & (oy < HH);
}

// ---- kernel 1: zero the winner-map at every position we may scatter to ---

__global__ void init_map_kernel(const int* __restrict__ inds_ref,
                                const int* __restrict__ inds_other,
                                unsigned* __restrict__ wmap) {
  int i = blockIdx.x * blockDim.x + threadIdx.x;
  if (i >= POINTS) return;
  int b = i / NPTS, n = i % NPTS;
  int rx, ry, ox, oy; bool valid;
  load_point(inds_ref, inds_other, b, n, rx, ry, ox, oy, valid);
  if (valid) {
    // NOTE: reference linearizes with stride H (faithful): lin = H*y + x
    int lin_o = HH * oy + ox;
    wmap[(size_t)b * HWSZ + lin_o] = 0u;
  }
}

// ---- kernel 2: per-point argmax gather + last-writer-wins scatter --------

__global__ void scatter_argmax_kernel(const float* __restrict__ inputs_ref,
                                      const int* __restrict__ inds_ref,
                                      const int* __restrict__ inds_other,
                                      unsigned* __restrict__ wmap) {
  int i = blockIdx.x * blockDim.x + threadIdx.x;
  if (i >= POINTS) return;
  int b = i / NPTS, n = i % NPTS;
  int rx, ry, ox, oy; bool valid;
  load_point(inds_ref, inds_other, b, n, rx, ry, ox, oy, valid);
  if (!valid) return;

  int lin_r = HH * ry + rx;
  lin_r = min(max(lin_r, 0), HWSZ - 1);   // faithful clip

  // argmax over the 19 channels at this single pixel (first-max tiebreak)
  const float* src = inputs_ref + (size_t)b * CCH * HWSZ + lin_r;
  float best = src[0];
  int cls = 0;
#pragma unroll
  for (int c = 1; c < CCH; ++c) {
    float v = src[(size_t)c * HWSZ];
    if (v > best) { best = v; cls = c; }
  }

  // sequential ".set" semantics: highest point index n wins at each pixel
  unsigned packed = ((unsigned)(n + 1) << 8) | (unsigned)cls;
  int lin_o = HH * oy + ox;
  atomicMax(&wmap[(size_t)b * HWSZ + lin_o], packed);
}

// ---- kernel 3: winner check + log-softmax pick, block partial reduce -----

__global__ void ce_partial_kernel(const float* __restrict__ inputs_other,
                                  const int* __restrict__ inds_ref,
                                  const int* __restrict__ inds_other,
                                  const unsigned* __restrict__ wmap,
                                  float* __restrict__ partials) {
  __shared__ float s_sum[TPB];
  __shared__ float s_cnt[TPB];

  int i = blockIdx.x * blockDim.x + threadIdx.x;
  float my_sum = 0.0f, my_cnt = 0.0f;

  if (i < POINTS) {
    int b = i / NPTS, n = i % NPTS;
    int rx, ry, ox, oy; bool valid;
    load_point(inds_ref, inds_other, b, n, rx, ry, ox, oy, valid);
    if (valid) {
      int lin_o = HH * oy + ox;
      unsigned v = wmap[(size_t)b * HWSZ + lin_o];
      if ((v >> 8) == (unsigned)(n + 1)) {       // this point owns the pixel
        int cls = (int)(v & 0xFFu);              // cls in [0,18] != IGNORE
        const float* x = inputs_other + (size_t)b * CCH * HWSZ + lin_o;
        float m = x[0];
#pragma unroll
        for (int c = 1; c < CCH; ++c) m = fmaxf(m, x[(size_t)c * HWSZ]);
        float s = 0.0f;
#pragma unroll
        for (int c = 0; c < CCH; ++c) s += expf(x[(size_t)c * HWSZ] - m);
        float lse = m + logf(s);
        my_sum = x[(size_t)cls * HWSZ] - lse;    // log-softmax picked value
        my_cnt = 1.0f;                           // one non-ignored pixel
      }
    }
  }

  s_sum[threadIdx.x] = my_sum;
  s_cnt[threadIdx.x] = my_cnt;
  __syncthreads();
  // fixed-order tree reduction: deterministic
  for (int off = TPB / 2; off > 0; off >>= 1) {
    if (threadIdx.x < off) {
      s_sum[threadIdx.x] += s_sum[threadIdx.x + off];
      s_cnt[threadIdx.x] += s_cnt[threadIdx.x + off];
    }
    __syncthreads();
  }
  if (threadIdx.x == 0) {
    partials[2 * blockIdx.x + 0] = s_sum[0];
    partials[2 * blockIdx.x + 1] = s_cnt[0];
  }
}

// ---- kernel 4: single-wave finisher; cross-lane sum via V_WMMA_F32_16X16X4_F32

__global__ void final_reduce_kernel(const float* __restrict__ partials,
                                    float* __restrict__ out) {
  int lane = threadIdx.x;            // 32 threads = one wave32
  float s = 0.0f, cnt = 0.0f;
  for (int i = lane; i < NBLK; i += 32) {  // fixed per-lane order: deterministic
    s   += partials[2 * i + 0];
    cnt += partials[2 * i + 1];
  }
  // Reconverged here -> EXEC all ones, legal for WMMA.
  // D = A(16x4, all ones) x B(4x16). B holds each lane's scalar exactly once
  // (b.y = 0 fills the other K slot), so every row of D is the 16 column
  // sums; summing c[0] over a 16-lane group with shfl_xor yields the total.
  v2f a;  a.x = 1.0f; a.y = 1.0f;
  v2f b1; b1.x = s;   b1.y = 0.0f;
  v2f b2; b2.x = cnt; b2.y = 0.0f;
  v8f c1 = {};
  v8f c2 = {};
  c1 = __builtin_amdgcn_wmma_f32_16x16x4_f32(false, a, false, b1, (short)0, c1, false, false);
  c2 = __builtin_amdgcn_wmma_f32_16x16x4_f32(false, a, false, b2, (short)0, c2, false, false);
  float cs = c1[0];
  float cc = c2[0];
#pragma unroll
  for (int off = 1; off < 16; off <<= 1) {   // 16 column sums -> total (wave32 shuffles)
    cs += __shfl_xor(cs, off, 32);
    cc += __shfl_xor(cc, off, 32);
  }
  if (lane == 0) out[0] = -cs / fmaxf(cc, 1.0f);
}

// ---- launch --------------------------------------------------------------

extern "C" void kernel_launch(void* const* d_in, const int* in_sizes, int n_in,
                              void* d_out, int out_size, void* d_ws, size_t ws_size,
                              hipStream_t stream) {
  const float* inputs_ref   = (const float*)d_in[0];   // [B,C,H,W] f32
  const float* inputs_other = (const float*)d_in[1];   // [B,C,H,W] f32
  const int*   inds_ref     = (const int*)d_in[2];     // [B,2,N]
  const int*   inds_other   = (const int*)d_in[3];     // [B,2,N]
  // d_in[4] (weights) is unused by the reference loss.

  unsigned* wmap   = (unsigned*)d_ws;                                   // B*HW u32
  float*    parts  = (float*)((char*)d_ws + (size_t)BATCH * HWSZ * 4);  // 2*NBLK f32

  init_map_kernel      <<<NBLK, TPB, 0, stream>>>(inds_ref, inds_other, wmap);
  scatter_argmax_kernel<<<NBLK, TPB, 0, stream>>>(inputs_ref, inds_ref, inds_other, wmap);
  ce_partial_kernel    <<<NBLK, TPB, 0, stream>>>(inputs_other, inds_ref, inds_other, wmap, parts);
  final_reduce_kernel  <<<1, 32, 0, stream>>>(parts, (float*)d_out);
}